// RAG4DyG_85529978732735
// MI455X (gfx1250) — compile-verified
//
#include <hip/hip_runtime.h>
#include <hip/hip_bf16.h>

typedef __attribute__((ext_vector_type(2))) float v2f;
typedef __attribute__((ext_vector_type(8))) float v8f;

// Problem constants (from setup_inputs)
static constexpr int NN = 100000;   // nodes
static constexpr int FF = 128;      // raw feature dim
static constexpr int PP = 50000;    // retrieval pool
static constexpr int LL = 100;      // sequence length
static constexpr int BB = 1024;     // batch
static constexpr int RR = 5;        // retrieved per sample
static constexpr int DD = 32;      // hidden dim
static constexpr int ROWS = BB * LL; // 102400, divisible by 16

static __device__ __forceinline__ v8f wmma4(v2f a, v2f b, v8f c) {
    return __builtin_amdgcn_wmma_f32_16x16x4_f32(
        false, a, false, b, (short)0, c, false, false);
}

// ---------------- K0: valid[b] = sum_l (demo_nodes[b,0,l] > 0) ----------------
__global__ __launch_bounds__(128) void k_valid(const int* __restrict__ rn,
                                               const int* __restrict__ ri,
                                               float* __restrict__ valid) {
    __shared__ int sred[4];
    int b = blockIdx.x;
    int tid = threadIdx.x;
    int p = ri[b * RR + 0];
    int c = 0;
    if (tid < LL) c = (rn[p * LL + tid] > 0) ? 1 : 0;
    #pragma unroll
    for (int off = 16; off >= 1; off >>= 1) c += __shfl_xor(c, off, 32);
    if ((tid & 31) == 0) sred[tid >> 5] = c;
    __syncthreads();
    if (tid == 0) valid[b] = (float)(sred[0] + sred[1] + sred[2] + sred[3]);
}

// ------- K1: fused gather(mean over R) + [16,128]x[128,32] WMMA + sim + LN -------
__global__ __launch_bounds__(128) void k_feat(
        const float* __restrict__ nrf, const int* __restrict__ rn,
        const int* __restrict__ ri, const int* __restrict__ dst_ids,
        const float* __restrict__ Wf, const float* __restrict__ bfeat,
        const float* __restrict__ wst, const float* __restrict__ bst,
        const float* __restrict__ lng, const float* __restrict__ lnb,
        float* __restrict__ hist) {
    __shared__ float As[16 * FF];   // averaged gathered features (A tile)
    __shared__ float Hs[16 * DD];   // projected tile before LN
    __shared__ float simS[16];

    int tile = blockIdx.x;
    int tid = threadIdx.x;           // 128 threads, f = tid
    // ---- gather + average over R, skill-match count ----
    for (int i = 0; i < 16; ++i) {
        int row = tile * 16 + i;
        int b = row / LL, l = row % LL;
        int dstskill = 0;
        if (tid == 0) dstskill = (int)nrf[(size_t)dst_ids[b] * FF];
        float acc = 0.f;
        int simc = 0;
        #pragma unroll
        for (int r = 0; r < RR; ++r) {
            int p = ri[b * RR + r];
            int node = rn[p * LL + l];
            float v = nrf[(size_t)node * FF + tid];
            acc += v;
            if (tid == 0) simc += ((int)v == dstskill) ? 1 : 0;
        }
        As[i * FF + tid] = acc * 0.2f;
        if (tid == 0) simS[i] = (float)simc * 0.2f;
    }
    __syncthreads();

    // ---- WMMA: waves 0/1 compute N-halves 0..15 / 16..31 ----
    int wave = tid >> 5, lane = tid & 31;
    if (wave < 2) {
        int ml = lane & 15;
        int khalf = lane >> 4;
        int n = wave * 16 + ml;
        v8f c = {};
        #pragma unroll
        for (int ks = 0; ks < FF / 4; ++ks) {
            int k0 = ks * 4 + khalf * 2;
            v2f a, bb;
            a.x = As[ml * FF + k0];
            a.y = As[ml * FF + k0 + 1];
            bb.x = Wf[k0 * DD + n];
            bb.y = Wf[(k0 + 1) * DD + n];
            c = wmma4(a, bb, c);
        }
        int nn = wave * 16 + (lane & 15);
        int mb = (lane >> 4) * 8;
        #pragma unroll
        for (int i = 0; i < 8; ++i) {
            int mm = mb + i;
            Hs[mm * DD + nn] = c[i] + bfeat[nn] + simS[mm] * wst[nn] + bst[nn];
        }
    }
    __syncthreads();

    // ---- LayerNorm over D=32: 4 waves x 4 rows, wave-wide reduction ----
    int ln = lane;
    #pragma unroll
    for (int rr = 0; rr < 4; ++rr) {
        int i = wave * 4 + rr;
        float x = Hs[i * DD + ln];
        float s1 = x, s2 = x * x;
        #pragma unroll
        for (int off = 16; off >= 1; off >>= 1) {
            s1 += __shfl_xor(s1, off, 32);
            s2 += __shfl_xor(s2, off, 32);
        }
        float mu = s1 * (1.f / 32.f);
        float var = s2 * (1.f / 32.f) - mu * mu;
        float y = (x - mu) * rsqrtf(var + 1e-5f) * lng[ln] + lnb[ln];
        int row = tile * 16 + i;
        hist[(size_t)row * DD + ln] = y;
    }
}

// ------- K2: per-sample GCN: xW (WMMA) + chain message + relu + mean pool -------
__global__ __launch_bounds__(128) void k_gcn(
        const float* __restrict__ hist, const float* __restrict__ Wg,
        const float* __restrict__ bg, const float* __restrict__ valid,
        float* __restrict__ srcpool) {
    __shared__ float Hs[112 * DD];
    __shared__ float Xs[112 * DD];
    __shared__ float pools[4 * DD];

    int b = blockIdx.x;
    int tid = threadIdx.x;
    for (int idx = tid; idx < 112 * DD; idx += 128)
        Hs[idx] = (idx < LL * DD) ? hist[(size_t)b * LL * DD + idx] : 0.f;
    __syncthreads();

    int wave = tid >> 5, lane = tid & 31;
    int ml = lane & 15, khalf = lane >> 4;
    // 7 M-tiles x 2 N-tiles = 14 tasks across 4 waves
    for (int t = wave; t < 14; t += 4) {
        int mtile = t >> 1, ntile = t & 1;
        v8f c = {};
        #pragma unroll
        for (int ks = 0; ks < DD / 4; ++ks) {
            int k0 = ks * 4 + khalf * 2;
            v2f a, bb;
            a.x = Hs[(mtile * 16 + ml) * DD + k0];
            a.y = Hs[(mtile * 16 + ml) * DD + k0 + 1];
            int n = ntile * 16 + ml;
            bb.x = Wg[k0 * DD + n];
            bb.y = Wg[(k0 + 1) * DD + n];
            c = wmma4(a, bb, c);
        }
        int nn = ntile * 16 + (lane & 15);
        int mb = (lane >> 4) * 8;
        #pragma unroll
        for (int i = 0; i < 8; ++i)
            Xs[(mtile * 16 + mb + i) * DD + nn] = c[i];
    }
    __syncthreads();

    float vb = valid[b];
    int n = tid & 31;
    int rl = tid >> 5;
    float pool = 0.f;
    for (int j = rl; j < LL; j += 4) {
        bool edge = (j >= 1) && ((float)j < vb);
        float degj = edge ? 2.f : 1.f;              // deg[0]==1; deg[j]=1+(j<valid)
        float out = Xs[j * DD + n] / degj;
        if (edge) {
            float degp = (j - 1 >= 1) ? 2.f : 1.f;  // edge implies (j-1)<valid
            out += Xs[(j - 1) * DD + n] * rsqrtf(degp * degj);
        }
        float g = out + bg[n];
        pool += (g > 0.f) ? g : 0.f;
    }
    pools[rl * DD + n] = pool;
    __syncthreads();
    if (tid < DD) {
        float s = pools[tid] + pools[DD + tid] + pools[2 * DD + tid] + pools[3 * DD + tid];
        srcpool[(size_t)b * DD + tid] = s * (1.f / (float)LL);
    }
}

// ------- K3: dst gather + [16,128]x[128,32] WMMA projection -------
__global__ __launch_bounds__(64) void k_dst(
        const float* __restrict__ nrf, const int* __restrict__ dst_ids,
        const float* __restrict__ Wf, const float* __restrict__ bfeat,
        float* __restrict__ dstfeat) {
    __shared__ float As[16 * FF];
    int tile = blockIdx.x;
    int tid = threadIdx.x; // 64
    for (int idx = tid; idx < 16 * FF; idx += 64) {
        int i = idx >> 7, f = idx & 127;
        int node = dst_ids[tile * 16 + i];
        As[idx] = nrf[(size_t)node * FF + f];
    }
    __syncthreads();
    int wave = tid >> 5, lane = tid & 31;
    int ml = lane & 15, khalf = lane >> 4;
    int n = wave * 16 + ml;
    v8f c = {};
    #pragma unroll
    for (int ks = 0; ks < FF / 4; ++ks) {
        int k0 = ks * 4 + khalf * 2;
        v2f a, bb;
        a.x = As[ml * FF + k0];
        a.y = As[ml * FF + k0 + 1];
        bb.x = Wf[k0 * DD + n];
        bb.y = Wf[(k0 + 1) * DD + n];
        c = wmma4(a, bb, c);
    }
    int nn = wave * 16 + (lane & 15);
    int mb = (lane >> 4) * 8;
    #pragma unroll
    for (int i = 0; i < 8; ++i) {
        int row = tile * 16 + mb + i;
        dstfeat[(size_t)row * DD + nn] = c[i] + bfeat[nn];
    }
}

// ------- K4: final [1024,32]x[32,32] WMMA for src (y=0) and dst (y=1) -------
__global__ __launch_bounds__(64) void k_out(
        const float* __restrict__ srcpool, const float* __restrict__ dstfeat,
        const float* __restrict__ Wo, const float* __restrict__ bo,
        float* __restrict__ out) {
    __shared__ float As[16 * DD];
    int tile = blockIdx.x;
    const float* src = (blockIdx.y == 0) ? srcpool : dstfeat;
    float* outp = out + (size_t)blockIdx.y * BB * DD;
    int tid = threadIdx.x; // 64
    for (int idx = tid; idx < 16 * DD; idx += 64)
        As[idx] = src[(size_t)tile * 16 * DD + idx];
    __syncthreads();
    int wave = tid >> 5, lane = tid & 31;
    int ml = lane & 15, khalf = lane >> 4;
    int n = wave * 16 + ml;
    v8f c = {};
    #pragma unroll
    for (int ks = 0; ks < DD / 4; ++ks) {
        int k0 = ks * 4 + khalf * 2;
        v2f a, bb;
        a.x = As[ml * DD + k0];
        a.y = As[ml * DD + k0 + 1];
        bb.x = Wo[k0 * DD + n];
        bb.y = Wo[(k0 + 1) * DD + n];
        c = wmma4(a, bb, c);
    }
    int nn = wave * 16 + (lane & 15);
    int mb = (lane >> 4) * 8;
    #pragma unroll
    for (int i = 0; i < 8; ++i) {
        int row = tile * 16 + mb + i;
        outp[(size_t)row * DD + nn] = c[i] + bo[nn];
    }
}

extern "C" void kernel_launch(void* const* d_in, const int* in_sizes, int n_in,
                              void* d_out, int out_size, void* d_ws, size_t ws_size,
                              hipStream_t stream) {
    (void)in_sizes; (void)n_in; (void)out_size; (void)ws_size;
    const float* nrf   = (const float*)d_in[0];
    const int*   rn    = (const int*)d_in[1];
    const int*   ri    = (const int*)d_in[2];
    // d_in[3] = src_node_ids (unused by reference outputs)
    const int*   dst   = (const int*)d_in[4];
    // d_in[5] = node_interact_times (unused)
    const float* Wf    = (const float*)d_in[6];
    const float* bfeat = (const float*)d_in[7];
    const float* wst   = (const float*)d_in[8];
    const float* bst   = (const float*)d_in[9];
    const float* lng   = (const float*)d_in[10];
    const float* lnb   = (const float*)d_in[11];
    const float* Wg    = (const float*)d_in[12];
    const float* bg    = (const float*)d_in[13];
    const float* Wo    = (const float*)d_in[14];
    const float* bo    = (const float*)d_in[15];

    float* ws = (float*)d_ws;
    float* hist    = ws;                       // ROWS*DD = 3,276,800 floats
    float* valid   = hist + (size_t)ROWS * DD; // BB floats
    float* srcpool = valid + BB;               // BB*DD
    float* dstfeat = srcpool + (size_t)BB * DD;

    k_valid<<<BB, 128, 0, stream>>>(rn, ri, valid);
    k_feat<<<ROWS / 16, 128, 0, stream>>>(nrf, rn, ri, dst, Wf, bfeat, wst, bst,
                                          lng, lnb, hist);
    k_gcn<<<BB, 128, 0, stream>>>(hist, Wg, bg, valid, srcpool);
    k_dst<<<BB / 16, 64, 0, stream>>>(nrf, dst, Wf, bfeat, dstfeat);
    k_out<<<dim3(BB / 16, 2), 64, 0, stream>>>(srcpool, dstfeat, Wo, bo,
                                               (float*)d_out);
}